// exp_LL_td_83545703841904
// MI455X (gfx1250) — compile-verified
//
#include <hip/hip_runtime.h>

typedef __attribute__((ext_vector_type(2))) float v2f;
typedef __attribute__((ext_vector_type(8))) float v8f;

#define DT_C 0.01f
#define KDIM 452   // 225 c_re + 225 c_im + 1 (bias/h_comm) + 1 pad
#define NDIM 256   // flattened 16x16 L

// ---------------------------------------------------------------------------
// Kernel 1: build constant weight W[452][256] such that
//   (u_s @ W) = flatten(DT * L_s),  u_s = [c_re(225), c_im(225), 1, 0]
// Folds M_re, M_im, tr_id map, h_comm, signs (-4/+4) and DT.
// ---------------------------------------------------------------------------
__global__ __launch_bounds__(256) void build_W_kernel(
    const float* __restrict__ f, const float* __restrict__ dm,
    const float* __restrict__ omega, float* __restrict__ W)
{
  int idx = blockIdx.x * 256 + threadIdx.x;
  if (idx >= KDIM * NDIM) return;
  int k = idx >> 8;          // 0..451  (input vector slot)
  int n = idx & 255;         // 0..255  (flattened L entry)
  int r = n >> 4, c = n & 15;
  float val = 0.0f;
  if (r > 0) {               // row 0 of L is zero
    int m = r - 1;
    if (k < 225) {           // c_re block -> -4*DT*M_re[m,nn,i,j]
      if (c > 0) {
        int nn = c - 1;
        int i = k / 15, j = k % 15;
        const float* fm = f + m * 225;
        const float* fn = f + nn * 225;
        float s = 0.0f;
        for (int kk = 0; kk < 15; ++kk)
          s += fm[j*15+kk]*fn[i*15+kk] + fm[i*15+kk]*fn[j*15+kk];
        val = -4.0f * DT_C * s;
      }
    } else if (k < 450) {    // c_im block
      int ij = k - 225;
      int i = ij / 15, j = ij % 15;
      if (c == 0) {          // tr_id column: -4*DT*f[i,m,j]
        val = -4.0f * DT_C * f[(i*15+m)*15+j];
      } else {               // +4*DT*M_im[m,nn,i,j]
        int nn = c - 1;
        const float* fm = f + m * 225;
        const float* dn = dm + nn * 225;
        float s = 0.0f;
        for (int kk = 0; kk < 15; ++kk)
          s += fm[i*15+kk]*dn[j*15+kk] - fm[j*15+kk]*dn[i*15+kk];
        val = 4.0f * DT_C * s;
      }
    } else if (k == 450) {   // constant-1 slot: DT*h_comm[m,nn] = 4*DT*sum f[nn,m,kk]*omega[kk]
      if (c > 0) {
        int nn = c - 1;
        float s = 0.0f;
        for (int kk = 0; kk < 15; ++kk)
          s += f[(nn*15+m)*15+kk] * omega[kk];
        val = 4.0f * DT_C * s;
      }
    }
    // k == 451: padding, stays 0
  }
  W[k*NDIM + n] = val;
}

// ---------------------------------------------------------------------------
// Kernel 2: fully fused per-16-sample pipeline.
//   Phase 0: MLPs -> vx,vy in LDS; Gram matrices -> U (16x452) in LDS
//   Phase 1: WMMA GEMM  Lflat(16x256) = U(16x452) @ W(452x256)   (f32 16x16x4)
//   Phase 2: per-wave expm(16x16) via scaling(2^-6) + order-8 Taylor Horner
//            + 6 squarings, all on V_WMMA_F32_16X16X4_F32; apply to [1;x].
// ---------------------------------------------------------------------------
__global__ __launch_bounds__(256) void fused_lindblad_kernel(
    const float* __restrict__ t, const float* __restrict__ x,
    const float* __restrict__ W,
    const float* __restrict__ wx1, const float* __restrict__ bx1,
    const float* __restrict__ wx2, const float* __restrict__ bx2,
    const float* __restrict__ wx3, const float* __restrict__ bx3,
    const float* __restrict__ wx4, const float* __restrict__ bx4,
    const float* __restrict__ wy1, const float* __restrict__ by1,
    const float* __restrict__ wy2, const float* __restrict__ by2,
    const float* __restrict__ wy3, const float* __restrict__ by3,
    const float* __restrict__ wy4, const float* __restrict__ by4,
    float* __restrict__ out)
{
  // Region A: vx/vy (16*450 floats); later reused: [0..4095]=Lbuf(16x256),
  //           [4096 + wave*256 .. ] = per-wave 16x16 staging tiles.
  __shared__ float VXY[16*450];
  __shared__ float U[16*KDIM];

  const int tid = threadIdx.x;
  const int sl  = tid >> 4;          // sample-in-block for phase 0 (0..15)
  const int r   = tid & 15;          // worker-in-sample (0..15)
  const int gs0 = blockIdx.x * 16;

  // ---------------- Phase 0a: tiny MLPs -> vx, vy ----------------
  {
    const float tv = t[gs0 + sl];
    const float h1x0 = fmaxf(fmaf(wx1[0], tv, bx1[0]), 0.0f);
    const float h1x1 = fmaxf(fmaf(wx1[1], tv, bx1[1]), 0.0f);
    const float h1y0 = fmaxf(fmaf(wy1[0], tv, by1[0]), 0.0f);
    const float h1y1 = fmaxf(fmaf(wy1[1], tv, by1[1]), 0.0f);
    float h2x[4], h2y[4];
    for (int o = 0; o < 4; ++o) {
      h2x[o] = fmaxf(wx2[o*2+0]*h1x0 + wx2[o*2+1]*h1x1 + bx2[o], 0.0f);
      h2y[o] = fmaxf(wy2[o*2+0]*h1y0 + wy2[o*2+1]*h1y1 + by2[o], 0.0f);
    }
    float h3x[4], h3y[4];
    for (int o = 0; o < 4; ++o) {
      float ax = bx3[o], ay = by3[o];
      for (int i = 0; i < 4; ++i) {
        ax += wx3[o*4+i]*h2x[i];
        ay += wy3[o*4+i]*h2y[i];
      }
      h3x[o] = fmaxf(ax, 0.0f);
      h3y[o] = fmaxf(ay, 0.0f);
    }
    float* vx = &VXY[sl*450];
    float* vy = vx + 225;
    for (int idx = r; idx < 225; idx += 16) {
      const float* w4x = wx4 + idx*4;
      const float* w4y = wy4 + idx*4;
      vx[idx] = w4x[0]*h3x[0] + w4x[1]*h3x[1] + w4x[2]*h3x[2] + w4x[3]*h3x[3] + bx4[idx];
      vy[idx] = w4y[0]*h3y[0] + w4y[1]*h3y[1] + w4y[2]*h3y[2] + w4y[3]*h3y[3] + by4[idx];
    }
  }
  __syncthreads();

  // ---------------- Phase 0b: Gram matrices -> U ----------------
  {
    const float* vx = &VXY[sl*450];
    const float* vy = vx + 225;
    float* Us = &U[sl*KDIM];
    for (int k = r; k < KDIM; k += 16) {
      float val;
      if (k < 225) {                       // c_re[i,j]
        const int i = k/15, j = k%15;
        float s = 0.0f;
        for (int kk = 0; kk < 15; ++kk)
          s += vx[kk*15+i]*vx[kk*15+j] + vy[kk*15+i]*vy[kk*15+j];
        val = s;
      } else if (k < 450) {                // c_im[i,j]
        const int ij = k - 225;
        const int i = ij/15, j = ij%15;
        float s = 0.0f;
        for (int kk = 0; kk < 15; ++kk)
          s += vx[kk*15+i]*vy[kk*15+j] - vy[kk*15+i]*vx[kk*15+j];
        val = s;
      } else {
        val = (k == 450) ? 1.0f : 0.0f;    // bias slot, pad slot
      }
      Us[k] = val;
    }
  }
  __syncthreads();

  // ---------------- Phase 1: WMMA GEMM (M=16 samples, N=256, K=452) --------
  const int wave  = tid >> 5;            // 0..7, handles N-tiles 2w, 2w+1
  const int lane  = tid & 31;
  const int lrow  = lane & 15;
  const int hi    = (lane >= 16) ? 1 : 0;
  const int koffA = hi ? 2 : 0;          // f32 A/B layout: low lanes K0,K1 / high K2,K3

  v8f acc0, acc1;
  for (int v = 0; v < 8; ++v) { acc0[v] = 0.0f; acc1[v] = 0.0f; }
  const int n0 = wave*32 + lrow;
  const int n1 = n0 + 16;
  for (int ks = 0; ks < KDIM/4; ++ks) {
    const int k0 = ks*4 + koffA;
    v2f a;                                // A: M = lane%16 (sample row)
    a.x = U[lrow*KDIM + k0];
    a.y = U[lrow*KDIM + k0 + 1];
    v2f b0, b1;                           // B: N = lane%16 within tile
    b0.x = W[ k0   *NDIM + n0];
    b0.y = W[(k0+1)*NDIM + n0];
    b1.x = W[ k0   *NDIM + n1];
    b1.y = W[(k0+1)*NDIM + n1];
    acc0 = __builtin_amdgcn_wmma_f32_16x16x4_f32(false, a, false, b0, (short)0, acc0, false, false);
    acc1 = __builtin_amdgcn_wmma_f32_16x16x4_f32(false, a, false, b1, (short)0, acc1, false, false);
  }

  // Scatter DT*L into LDS (reuse VXY region): Lbuf[sample*256 + n]
  float* Lbuf = VXY;
  const int srow = hi ? 8 : 0;           // C layout: M = vgpr + (hi?8:0)
  for (int v = 0; v < 8; ++v) {
    Lbuf[(v + srow)*256 + n0] = acc0[v];
    Lbuf[(v + srow)*256 + n1] = acc1[v];
  }
  __syncthreads();

  // ---------------- Phase 2: expm per wave (2 samples each) ----------------
  float* STG = VXY + 4096 + wave*256;    // private 16x16 staging tile
  for (int si = 0; si < 2; ++si) {
    const int s = wave*2 + si;
    const float* Ls = Lbuf + s*256;

    // A-layout fragments of Bm = (DT*L) * 2^-6, kept in regs for all iters
    v2f af[4];
    for (int kb = 0; kb < 4; ++kb) {
      af[kb].x = Ls[lrow*16 + kb*4 + koffA    ] * 0.015625f;
      af[kb].y = Ls[lrow*16 + kb*4 + koffA + 1] * 0.015625f;
    }

    // P = I (C layout)
    v8f P;
    for (int v = 0; v < 8; ++v) P[v] = ((v + srow) == lrow) ? 1.0f : 0.0f;

    // Matrix Horner: P <- I + (Bm/kq)*P, kq = 8..1  => order-8 Taylor of exp(Bm)
    for (int kq = 8; kq >= 1; --kq) {
      const float inv = 1.0f / (float)kq;
      for (int v = 0; v < 8; ++v) STG[(v + srow)*16 + lrow] = P[v]*inv;
      v8f acc;
      for (int v = 0; v < 8; ++v) acc[v] = ((v + srow) == lrow) ? 1.0f : 0.0f;
      for (int kb = 0; kb < 4; ++kb) {
        v2f b;
        b.x = STG[(kb*4 + koffA    )*16 + lrow];
        b.y = STG[(kb*4 + koffA + 1)*16 + lrow];
        acc = __builtin_amdgcn_wmma_f32_16x16x4_f32(false, af[kb], false, b, (short)0, acc, false, false);
      }
      P = acc;
    }

    // 6 squarings: exp(DT*L) = (exp(Bm))^(2^6)
    for (int sq = 0; sq < 6; ++sq) {
      for (int v = 0; v < 8; ++v) STG[(v + srow)*16 + lrow] = P[v];
      v2f a2[4];
      for (int kb = 0; kb < 4; ++kb) {
        a2[kb].x = STG[lrow*16 + kb*4 + koffA    ];
        a2[kb].y = STG[lrow*16 + kb*4 + koffA + 1];
      }
      v8f acc;
      for (int v = 0; v < 8; ++v) acc[v] = 0.0f;
      for (int kb = 0; kb < 4; ++kb) {
        v2f b;
        b.x = STG[(kb*4 + koffA    )*16 + lrow];
        b.y = STG[(kb*4 + koffA + 1)*16 + lrow];
        acc = __builtin_amdgcn_wmma_f32_16x16x4_f32(false, a2[kb], false, b, (short)0, acc, false, false);
      }
      P = acc;
    }

    // Apply: out[s,j] = E[1+j,0] + sum_i x[s,i]*E[1+j,1+i]
    for (int v = 0; v < 8; ++v) STG[(v + srow)*16 + lrow] = P[v];
    if (lane < 15) {
      const int gsamp = gs0 + s;
      const float* xs = x + gsamp*15;
      const float* Er = STG + (1 + lane)*16;
      float accv = Er[0];
      for (int i = 0; i < 15; ++i) accv = fmaf(xs[i], Er[1+i], accv);
      out[gsamp*15 + lane] = accv;
    }
  }
}

// ---------------------------------------------------------------------------
extern "C" void kernel_launch(void* const* d_in, const int* in_sizes, int n_in,
                              void* d_out, int out_size, void* d_ws, size_t ws_size,
                              hipStream_t stream) {
  const float* t     = (const float*)d_in[0];
  const float* x     = (const float*)d_in[1];
  const float* omega = (const float*)d_in[2];
  const float* f     = (const float*)d_in[3];
  const float* dm    = (const float*)d_in[4];
  const float* wx1 = (const float*)d_in[5];
  const float* bx1 = (const float*)d_in[6];
  const float* wx2 = (const float*)d_in[7];
  const float* bx2 = (const float*)d_in[8];
  const float* wx3 = (const float*)d_in[9];
  const float* bx3 = (const float*)d_in[10];
  const float* wx4 = (const float*)d_in[11];
  const float* bx4 = (const float*)d_in[12];
  const float* wy1 = (const float*)d_in[13];
  const float* by1 = (const float*)d_in[14];
  const float* wy2 = (const float*)d_in[15];
  const float* by2 = (const float*)d_in[16];
  const float* wy3 = (const float*)d_in[17];
  const float* by3 = (const float*)d_in[18];
  const float* wy4 = (const float*)d_in[19];
  const float* by4 = (const float*)d_in[20];
  float* W   = (float*)d_ws;               // 452*256 floats = 463 KB
  float* out = (float*)d_out;
  const int B = in_sizes[0];               // 65536

  build_W_kernel<<<(KDIM*NDIM + 255)/256, 256, 0, stream>>>(f, dm, omega, W);
  fused_lindblad_kernel<<<B/16, 256, 0, stream>>>(
      t, x, W, wx1,bx1,wx2,bx2,wx3,bx3,wx4,bx4,
      wy1,by1,wy2,by2,wy3,by3,wy4,by4, out);
}